// Simulator_23416161698037
// MI455X (gfx1250) — compile-verified
//
#include <hip/hip_runtime.h>
#include <hip/hip_bf16.h>

#define N_NODES 10000
#define N_EDGES 70000
#define H 128
#define L_BLOCKS 15
#define TILE 16
#define ROWS 64            // rows per block in the big GEMM kernels (4 row-tiles/wave)
#define LN_EPS 1e-5f

typedef __attribute__((ext_vector_type(2))) float v2f;
typedef __attribute__((ext_vector_type(8))) float v8f;

// ---------- WMMA fp32 16x16x4 fragment helpers (wave32) ----------
// A 16x4 (MxK): lanes 0-15 m=lane, V0=K0,V1=K1 ; lanes 16-31: V0=K2,V1=K3
__device__ __forceinline__ v2f lda_frag(const float* A, int lda, int k0, int lane) {
    const int m = lane & 15;
    const int h2 = (lane >> 4) << 1;
    v2f a;
    a.x = A[m * lda + k0 + h2];
    a.y = A[m * lda + k0 + h2 + 1];
    return a;
}
// B 4x16 (KxN), W row-major [K][ldw], tile columns [col, col+16)
__device__ __forceinline__ v2f ldb_frag(const float* W, int ldw, int k0, int col, int lane) {
    const int n = lane & 15;
    const int h2 = (lane >> 4) << 1;
    v2f b;
    b.x = W[(k0 + h2) * ldw + col + n];
    b.y = W[(k0 + h2 + 1) * ldw + col + n];
    return b;
}
// guarded variant for zero-padded K (encoder MLPs with K=11 / K=3)
__device__ __forceinline__ v2f ldb_frag_guard(const float* W, int ldw, int k0, int col,
                                              int lane, int Kreal) {
    const int n = lane & 15;
    const int h2 = (lane >> 4) << 1;
    const int k1 = k0 + h2, k2 = k0 + h2 + 1;
    v2f b;
    b.x = (k1 < Kreal) ? W[k1 * ldw + col + n] : 0.0f;
    b.y = (k2 < Kreal) ? W[k2 * ldw + col + n] : 0.0f;
    return b;
}
__device__ __forceinline__ v8f wmma4(v2f a, v2f b, v8f c) {
    return __builtin_amdgcn_wmma_f32_16x16x4_f32(false, a, false, b, (short)0, c,
                                                 false, false);
}
// single 16-row tile: D = A(16xK LDS) @ W(KxH global, cols [col,col+16))
__device__ __forceinline__ v8f gemm_tile(const float* A, int lda, const float* W,
                                         int col, int lane, int K) {
    v8f acc = {};
#pragma unroll 4
    for (int k = 0; k < K; k += 4)
        acc = wmma4(lda_frag(A, lda, k, lane), ldb_frag(W, H, k, col, lane), acc);
    return acc;
}
// four 16-row tiles sharing one B-fragment load per k-step (4x weight reuse)
struct Acc4 { v8f a0, a1, a2, a3; };
__device__ __forceinline__ Acc4 gemm_tile4(const float* A, int lda, const float* W,
                                           int col, int lane, int K) {
    v8f a0 = {};
    v8f a1 = {};
    v8f a2 = {};
    v8f a3 = {};
#pragma unroll 4
    for (int k = 0; k < K; k += 4) {
        v2f b = ldb_frag(W, H, k, col, lane);
        a0 = wmma4(lda_frag(A, lda, k, lane), b, a0);
        a1 = wmma4(lda_frag(A + 16 * lda, lda, k, lane), b, a1);
        a2 = wmma4(lda_frag(A + 32 * lda, lda, k, lane), b, a2);
        a3 = wmma4(lda_frag(A + 48 * lda, lda, k, lane), b, a3);
    }
    Acc4 r;
    r.a0 = a0; r.a1 = a1; r.a2 = a2; r.a3 = a3;
    return r;
}
// store 4 row-tiles (+bias, optional relu) into LDS with row stride ldd
__device__ __forceinline__ void store_acc4(float* dst, int ldd, const Acc4& a,
                                           int col0, int lane, float bias, bool relu) {
    const int n = lane & 15, hh = lane >> 4;
#pragma unroll
    for (int r = 0; r < 8; ++r) {
        float v0 = a.a0[r] + bias;
        float v1 = a.a1[r] + bias;
        float v2 = a.a2[r] + bias;
        float v3 = a.a3[r] + bias;
        if (relu) {
            v0 = v0 > 0.f ? v0 : 0.f;
            v1 = v1 > 0.f ? v1 : 0.f;
            v2 = v2 > 0.f ? v2 : 0.f;
            v3 = v3 > 0.f ? v3 : 0.f;
        }
        const int row = r + 8 * hh;
        dst[(row) * ldd + col0 + n] = v0;
        dst[(row + 16) * ldd + col0 + n] = v1;
        dst[(row + 32) * ldd + col0 + n] = v2;
        dst[(row + 48) * ldd + col0 + n] = v3;
    }
}

// ---------- row LayerNorm stats over sOut[R][ldo], one thread per row ----------
__device__ __forceinline__ void ln_stats(const float* sOut, int ldo, float* sMu,
                                         float* sRs, int t, int nrows) {
    if (t < nrows) {
        float s = 0.f, ss = 0.f;
#pragma unroll 4
        for (int c = 0; c < H; ++c) {
            float v = sOut[t * ldo + c];
            s += v;
            ss += v * v;
        }
        float mu = s * (1.0f / H);
        float var = ss * (1.0f / H) - mu * mu;
        sMu[t] = mu;
        sRs[t] = rsqrtf(var + LN_EPS);
    }
}

// =================== encoder: node MLP (11 -> 128 -> 128, LN) ===================
__global__ __launch_bounds__(256) void enc_node_kernel(
    const int* __restrict__ node_type, const float* __restrict__ frames,
    const float* __restrict__ nm, const float* __restrict__ ns,
    const float* __restrict__ W1, const float* __restrict__ b1,
    const float* __restrict__ W2, const float* __restrict__ b2,
    const float* __restrict__ g, const float* __restrict__ bb,
    float* __restrict__ x) {
    __shared__ float sIn[TILE * 16];
    __shared__ float sHid[TILE * H];
    __shared__ float sOut[TILE * H];
    __shared__ float sMu[TILE], sRs[TILE];

    const int t = threadIdx.x;
    const int lane = t & 31;
    const int wave = t >> 5;
    const int base = blockIdx.x * TILE;

    {   // build normalized node_attr tile [16][11] padded to K=16
        const int i = t >> 4, k = t & 15;
        const int node = base + i;
        float feat;
        if (k < 2) feat = frames[node * 2 + k];
        else if (k < 11) feat = (node_type[node] == (k - 2)) ? 1.0f : 0.0f;
        else feat = 0.0f;
        sIn[i * 16 + k] = (k < 11) ? (feat - nm[k]) / ns[k] : 0.0f;
    }
    __syncthreads();

    {   // hid = relu(attr @ W1 + b1)  (K padded; W rows >= 11 read as zero)
        const int col0 = wave * 16;
        v8f acc = {};
#pragma unroll
        for (int k = 0; k < 16; k += 4)
            acc = wmma4(lda_frag(sIn, 16, k, lane),
                        ldb_frag_guard(W1, H, k, col0, lane, 11), acc);
        const int n = lane & 15, hh = lane >> 4;
#pragma unroll
        for (int r = 0; r < 8; ++r) {
            float v = acc[r] + b1[col0 + n];
            sHid[(r + 8 * hh) * H + col0 + n] = v > 0.f ? v : 0.f;
        }
    }
    __syncthreads();

    {   // out = hid @ W2 + b2
        const int col0 = wave * 16;
        v8f acc = gemm_tile(sHid, H, W2, col0, lane, H);
        const int n = lane & 15, hh = lane >> 4;
#pragma unroll
        for (int r = 0; r < 8; ++r)
            sOut[(r + 8 * hh) * H + col0 + n] = acc[r] + b2[col0 + n];
    }
    __syncthreads();

    ln_stats(sOut, H, sMu, sRs, t, TILE);
    __syncthreads();

    {   // x = LN(out)*g + b
        const int i = t >> 4, l16 = t & 15;
#pragma unroll
        for (int j = 0; j < 8; ++j) {
            const int c = l16 + 16 * j;
            float v = (sOut[i * H + c] - sMu[i]) * sRs[i] * g[c] + bb[c];
            x[(size_t)(base + i) * H + c] = v;
        }
    }
}

// =================== encoder: edge MLP (3 -> 128 -> 128, LN) ===================
__global__ __launch_bounds__(256) void enc_edge_kernel(
    const float* __restrict__ ef,
    const float* __restrict__ W1, const float* __restrict__ b1,
    const float* __restrict__ W2, const float* __restrict__ b2,
    const float* __restrict__ g, const float* __restrict__ bb,
    float* __restrict__ e) {
    __shared__ float sIn[TILE * 4];
    __shared__ float sHid[TILE * H];
    __shared__ float sOut[TILE * H];
    __shared__ float sMu[TILE], sRs[TILE];

    const int t = threadIdx.x;
    const int lane = t & 31;
    const int wave = t >> 5;
    const int base = blockIdx.x * TILE;

    if (t < 64) {
        const int i = t >> 2, k = t & 3;
        sIn[i * 4 + k] = (k < 3) ? ef[(base + i) * 3 + k] : 0.0f;
    }
    __syncthreads();

    {
        const int col0 = wave * 16;
        v8f acc = {};
        acc = wmma4(lda_frag(sIn, 4, 0, lane),
                    ldb_frag_guard(W1, H, 0, col0, lane, 3), acc);
        const int n = lane & 15, hh = lane >> 4;
#pragma unroll
        for (int r = 0; r < 8; ++r) {
            float v = acc[r] + b1[col0 + n];
            sHid[(r + 8 * hh) * H + col0 + n] = v > 0.f ? v : 0.f;
        }
    }
    __syncthreads();

    {
        const int col0 = wave * 16;
        v8f acc = gemm_tile(sHid, H, W2, col0, lane, H);
        const int n = lane & 15, hh = lane >> 4;
#pragma unroll
        for (int r = 0; r < 8; ++r)
            sOut[(r + 8 * hh) * H + col0 + n] = acc[r] + b2[col0 + n];
    }
    __syncthreads();

    ln_stats(sOut, H, sMu, sRs, t, TILE);
    __syncthreads();

    {
        const int i = t >> 4, l16 = t & 15;
#pragma unroll
        for (int j = 0; j < 8; ++j) {
            const int c = l16 + 16 * j;
            float v = (sOut[i * H + c] - sMu[i]) * sRs[i] * g[c] + bb[c];
            e[(size_t)(base + i) * H + c] = v;
        }
    }
}

// ========= edge block (64 edges/block, 4 row-tiles/wave):
//   e_new = MLP_LN(concat(x[s],x[r],e)); e+=e_new; agg[recv]+=e_new
//   LDS: sIn [64][384] (cols 0..127 reused as the out tile after stage 1), sHid [64][128]
__global__ __launch_bounds__(256) void edge_block_kernel(
    const float* __restrict__ x, float* __restrict__ e, float* __restrict__ agg,
    const int* __restrict__ senders, const int* __restrict__ receivers,
    const float* __restrict__ W1, const float* __restrict__ b1,
    const float* __restrict__ W2, const float* __restrict__ b2,
    const float* __restrict__ g, const float* __restrict__ bb) {
    __shared__ float sIn[ROWS * 3 * H];   // 96 KB: [x_s | x_r | e]; out overlays cols 0..127
    __shared__ float sHid[ROWS * H];      // 32 KB
    __shared__ float sMu[ROWS], sRs[ROWS];
    __shared__ int sRecv[ROWS];

    const int t = threadIdx.x;
    const int lane = t & 31;
    const int wave = t >> 5;
    const int ebase = blockIdx.x * ROWS;

    // warm L2/L1 with this layer's weights while we gather
    __builtin_prefetch(W1 + t * 192, 0, 0);   // covers 384*128 floats
    __builtin_prefetch(W2 + t * 64, 0, 0);    // covers 128*128 floats

    if (t < ROWS) {
        const int row = ebase + t;
        sRecv[t] = receivers[row < N_EDGES ? row : N_EDGES - 1];
    }
    {   // gather concat tile: 4 threads per edge row, 24 float4 each
        const int i = t >> 2, l4 = t & 3;
        const int row = ebase + i;
        const int eidx = row < N_EDGES ? row : N_EDGES - 1;  // clamp tail (discarded)
        const int s = senders[eidx];
        const int r = receivers[eidx];
        const float4* xs = (const float4*)(x + (size_t)s * H);
        const float4* xr = (const float4*)(x + (size_t)r * H);
        const float4* ee = (const float4*)(e + (size_t)eidx * H);
        float4* dst = (float4*)(&sIn[i * 3 * H]);
#pragma unroll
        for (int j = 0; j < 24; ++j) {
            const int f4 = l4 + 4 * j;  // 0..95
            dst[f4] = (f4 < 32) ? xs[f4] : (f4 < 64) ? xr[f4 - 32] : ee[f4 - 64];
        }
    }
    __syncthreads();

    {   // hid = relu(in @ W1 + b1), K = 384, four row-tiles per wave
        const int col0 = wave * 16;
        Acc4 a = gemm_tile4(sIn, 3 * H, W1, col0, lane, 3 * H);
        store_acc4(sHid, H, a, col0, lane, b1[col0 + (lane & 15)], true);
    }
    __syncthreads();   // sIn cols 0..255 now dead; safe to overlay out tile

    {   // out = hid @ W2 + b2 -> overlay into sIn cols 0..127 (row stride 384)
        const int col0 = wave * 16;
        Acc4 a = gemm_tile4(sHid, H, W2, col0, lane, H);
        store_acc4(sIn, 3 * H, a, col0, lane, b2[col0 + (lane & 15)], false);
    }
    __syncthreads();

    ln_stats(sIn, 3 * H, sMu, sRs, t, ROWS);
    __syncthreads();

    {   // e += e_new ; agg[recv] += e_new (L2-resident float atomics)
        const int i = t >> 2, l4 = t & 3;
        if (ebase + i < N_EDGES) {
#pragma unroll
            for (int j = 0; j < 32; ++j) {
                const int c = l4 + 4 * j;
                float v = (sIn[i * 3 * H + c] - sMu[i]) * sRs[i] * g[c] + bb[c];
                e[(size_t)(ebase + i) * H + c] = sIn[i * 3 * H + 2 * H + c] + v;
                atomicAdd(&agg[(size_t)sRecv[i] * H + c], v);
            }
        }
    }
}

// ========= node block (64 nodes/block): x += MLP_LN(concat(x, agg))
//   LDS: sIn [64][256] (agg half reused as out tile after stage 1), sHid [64][128]
__global__ __launch_bounds__(256) void node_block_kernel(
    float* __restrict__ x, const float* __restrict__ agg,
    const float* __restrict__ W1, const float* __restrict__ b1,
    const float* __restrict__ W2, const float* __restrict__ b2,
    const float* __restrict__ g, const float* __restrict__ bb) {
    __shared__ float sIn[ROWS * 2 * H];   // 64 KB: [x | agg]; out overlays cols 128..255
    __shared__ float sHid[ROWS * H];      // 32 KB
    __shared__ float sMu[ROWS], sRs[ROWS];

    const int t = threadIdx.x;
    const int lane = t & 31;
    const int wave = t >> 5;
    const int base = blockIdx.x * ROWS;

    __builtin_prefetch(W1 + t * 128, 0, 0);
    __builtin_prefetch(W2 + t * 64, 0, 0);

    {   // 4 threads per node row, 16 float4 each
        const int i = t >> 2, l4 = t & 3;
        const int row = base + i;
        const int nidx = row < N_NODES ? row : N_NODES - 1;  // clamp tail (discarded)
        const float4* xn = (const float4*)(x + (size_t)nidx * H);
        const float4* an = (const float4*)(agg + (size_t)nidx * H);
        float4* dst = (float4*)(&sIn[i * 2 * H]);
#pragma unroll
        for (int j = 0; j < 16; ++j) {
            const int f4 = l4 + 4 * j;  // 0..63
            dst[f4] = (f4 < 32) ? xn[f4] : an[f4 - 32];
        }
    }
    __syncthreads();

    {
        const int col0 = wave * 16;
        Acc4 a = gemm_tile4(sIn, 2 * H, W1, col0, lane, 2 * H);
        store_acc4(sHid, H, a, col0, lane, b1[col0 + (lane & 15)], true);
    }
    __syncthreads();   // agg half of sIn now dead

    {   // out overlays sIn cols 128..255 (row stride 256)
        const int col0 = wave * 16;
        Acc4 a = gemm_tile4(sHid, H, W2, col0, lane, H);
        store_acc4(sIn + H, 2 * H, a, col0, lane, b2[col0 + (lane & 15)], false);
    }
    __syncthreads();

    ln_stats(sIn + H, 2 * H, sMu, sRs, t, ROWS);
    __syncthreads();

    {
        const int i = t >> 2, l4 = t & 3;
        if (base + i < N_NODES) {
#pragma unroll
            for (int j = 0; j < 32; ++j) {
                const int c = l4 + 4 * j;
                float v = (sIn[i * 2 * H + H + c] - sMu[i]) * sRs[i] * g[c] + bb[c];
                x[(size_t)(base + i) * H + c] = sIn[i * 2 * H + c] + v;  // residual
            }
        }
    }
}

// ========= decoder: out = frames + (relu(x@W1+b1)@W2+b2)*std + mean =========
__global__ __launch_bounds__(128) void decoder_kernel(
    const float* __restrict__ x, const float* __restrict__ frames,
    const float* __restrict__ W1, const float* __restrict__ b1,
    const float* __restrict__ W2, const float* __restrict__ b2,
    const float* __restrict__ om, const float* __restrict__ os,
    float* __restrict__ out) {
    __shared__ float sx[H];
    __shared__ float sh[H];
    const int node = blockIdx.x;
    const int t = threadIdx.x;
    sx[t] = x[(size_t)node * H + t];
    __syncthreads();
    float acc = b1[t];
#pragma unroll 8
    for (int k = 0; k < H; ++k) acc += sx[k] * W1[k * H + t];
    sh[t] = acc > 0.f ? acc : 0.f;
    __syncthreads();
    if (t < 2) {
        float o = b2[t];
#pragma unroll 8
        for (int k = 0; k < H; ++k) o += sh[k] * W2[k * 2 + t];
        out[node * 2 + t] = frames[node * 2 + t] + o * os[t] + om[t];
    }
}

extern "C" void kernel_launch(void* const* d_in, const int* in_sizes, int n_in,
                              void* d_out, int out_size, void* d_ws, size_t ws_size,
                              hipStream_t stream) {
    const int* node_type = (const int*)d_in[0];
    const int* edge_index = (const int*)d_in[1];
    const float* frames = (const float*)d_in[2];
    const float* edge_feats = (const float*)d_in[3];
    const float* node_nm = (const float*)d_in[4];
    const float* node_ns = (const float*)d_in[5];
    const float* out_nm = (const float*)d_in[6];
    const float* out_ns = (const float*)d_in[7];
    const float* enc_nW1 = (const float*)d_in[8];
    const float* enc_nb1 = (const float*)d_in[9];
    const float* enc_nW2 = (const float*)d_in[10];
    const float* enc_nb2 = (const float*)d_in[11];
    const float* enc_nlng = (const float*)d_in[12];
    const float* enc_nlnb = (const float*)d_in[13];
    const float* enc_eW1 = (const float*)d_in[14];
    const float* enc_eb1 = (const float*)d_in[15];
    const float* enc_eW2 = (const float*)d_in[16];
    const float* enc_eb2 = (const float*)d_in[17];
    const float* enc_elng = (const float*)d_in[18];
    const float* enc_elnb = (const float*)d_in[19];
    const float* eb_W1 = (const float*)d_in[20];
    const float* eb_b1 = (const float*)d_in[21];
    const float* eb_W2 = (const float*)d_in[22];
    const float* eb_b2 = (const float*)d_in[23];
    const float* eb_lng = (const float*)d_in[24];
    const float* eb_lnb = (const float*)d_in[25];
    const float* nb_W1 = (const float*)d_in[26];
    const float* nb_b1 = (const float*)d_in[27];
    const float* nb_W2 = (const float*)d_in[28];
    const float* nb_b2 = (const float*)d_in[29];
    const float* nb_lng = (const float*)d_in[30];
    const float* nb_lnb = (const float*)d_in[31];
    const float* dec_W1 = (const float*)d_in[32];
    const float* dec_b1 = (const float*)d_in[33];
    const float* dec_W2 = (const float*)d_in[34];
    const float* dec_b2 = (const float*)d_in[35];

    const int* senders = edge_index;             // edge_index[0]
    const int* receivers = edge_index + N_EDGES; // edge_index[1]

    float* x = (float*)d_ws;                      // [N, H]
    float* e = x + (size_t)N_NODES * H;           // [E, H]
    float* agg = e + (size_t)N_EDGES * H;         // [N, H]

    const int eb_grid = (N_EDGES + ROWS - 1) / ROWS;  // 1094
    const int nb_grid = (N_NODES + ROWS - 1) / ROWS;  // 157

    enc_node_kernel<<<N_NODES / TILE, 256, 0, stream>>>(
        node_type, frames, node_nm, node_ns, enc_nW1, enc_nb1, enc_nW2, enc_nb2,
        enc_nlng, enc_nlnb, x);
    enc_edge_kernel<<<N_EDGES / TILE, 256, 0, stream>>>(
        edge_feats, enc_eW1, enc_eb1, enc_eW2, enc_eb2, enc_elng, enc_elnb, e);

    for (int l = 0; l < L_BLOCKS; ++l) {
        hipMemsetAsync(agg, 0, (size_t)N_NODES * H * sizeof(float), stream);
        edge_block_kernel<<<eb_grid, 256, 0, stream>>>(
            x, e, agg, senders, receivers,
            eb_W1 + (size_t)l * 3 * H * H, eb_b1 + (size_t)l * H,
            eb_W2 + (size_t)l * H * H, eb_b2 + (size_t)l * H,
            eb_lng + (size_t)l * H, eb_lnb + (size_t)l * H);
        node_block_kernel<<<nb_grid, 256, 0, stream>>>(
            x, agg,
            nb_W1 + (size_t)l * 2 * H * H, nb_b1 + (size_t)l * H,
            nb_W2 + (size_t)l * H * H, nb_b2 + (size_t)l * H,
            nb_lng + (size_t)l * H, nb_lnb + (size_t)l * H);
    }

    decoder_kernel<<<N_NODES, 128, 0, stream>>>(
        x, frames, dec_W1, dec_b1, dec_W2, dec_b2, out_nm, out_ns, (float*)d_out);
}